// InterConv_9904194585366
// MI455X (gfx1250) — compile-verified
//
#include <hip/hip_runtime.h>

typedef __attribute__((ext_vector_type(2))) float v2f;
typedef __attribute__((ext_vector_type(4))) float v4f;
typedef __attribute__((ext_vector_type(8))) float v8f;

#define Bn 2048
#define Fn 39
#define En 64
#define Cn 64
#define Pn 741              // F*(F-1)/2
#define OUT_PER_B (Cn * Pn) // 47424
#define XST 68              // X stage stride (floats): 8B-aligned pairs, bank-conflict-free
#define AST 65              // result stage stride (floats): kills column-read conflicts
#define MROWS 48            // M padded to 3 tiles of 16

__global__ __launch_bounds__(256) void interconv_kernel(
    const float* __restrict__ x,     // (B, F, E)
    const float* __restrict__ W,     // (C, 1, 2, E)
    const float* __restrict__ bias,  // (C,)
    float* __restrict__ out)         // (B, C*P) flat
{
    __shared__ float Xs[MROWS * XST];           // staged X[b], zero-padded rows
    __shared__ float As[2][MROWS * AST];        // A = X*Wi^T, B2 = X*Wj^T
    __shared__ unsigned char iiL[Pn], jjL[Pn];  // pair lookup tables

    const int tid = threadIdx.x;
    const int b   = blockIdx.x;
    const int ln  = tid & 31;   // lane in wave32
    const int wv  = tid >> 5;   // wave id (0..7)

    // ---- build triu(k=1) pair LUT, once per block ----
    for (int p = tid; p < Pn; p += 256) {
        int i = 0, s = 0;
        while (s + (Fn - 1 - i) <= p) { s += (Fn - 1 - i); ++i; }
        iiL[p] = (unsigned char)i;
        jjL[p] = (unsigned char)(i + 1 + (p - s));
    }

    // ---- stage X[b] into LDS (coalesced float4), zero-pad rows 39..47 ----
    {
        const v4f* xb4 = (const v4f*)(x + (size_t)b * Fn * En);
        for (int idx = tid; idx < Fn * (En / 4); idx += 256) {  // 624 float4s
            const int row = idx >> 4;
            const int c4  = (idx & 15) << 2;
            const v4f v = xb4[idx];
            *(v4f*)&Xs[row * XST + c4] = v;
        }
        const v4f z = {};
        for (int idx = tid; idx < (MROWS - Fn) * (En / 4); idx += 256) {
            const int row = Fn + (idx >> 4);
            const int c4  = (idx & 15) << 2;
            *(v4f*)&Xs[row * XST + c4] = z;
        }
    }
    __syncthreads();

    // ---- phase 1: two 48x64 GEMMs via V_WMMA_F32_16X16X4_F32 ----
    const int m16 = ln & 15;   // row/col within 16-tile
    const int hi  = ln >> 4;   // lane half selects K pair (per ISA layout)

    // 24 tile jobs: {w in 0..1} x {m_tile 0..2} x {n_tile 0..3}
    for (int job = wv; job < 24; job += 8) {
        const int w   = job / 12;
        const int rem = job % 12;
        const int mt  = rem >> 2;
        const int nt  = rem & 3;

        const int row = mt * 16 + m16;          // M index into staged X
        const int col = nt * 16 + m16;          // N index = output channel c
        const float* xrow = &Xs[row * XST];     // LDS, zero-padded: no predication
        const float* wrow = W + (size_t)col * (2 * En) + (size_t)w * En;

        v8f acc = {};
        #pragma unroll 4
        for (int kk = 0; kk < 16; ++kk) {
            const int k = kk * 4 + 2 * hi;      // this lane's K pair
            const v2f a  = *(const v2f*)&xrow[k];   // ds_load_b64
            v2f bb;
            bb.x = wrow[k];
            bb.y = wrow[k + 1];                     // global_load_b64 (L2-resident)
            // D = A(16x4,f32) x B(4x16,f32) + C(16x16,f32)
            acc = __builtin_amdgcn_wmma_f32_16x16x4_f32(
                false, a, false, bb, (short)0, acc, false, false);
        }

        // scatter D tile to padded LDS: D row = v + 8*hi per ISA C/D layout
        float* dst = &As[w][0];
        #pragma unroll
        for (int v = 0; v < 8; ++v)
            dst[(mt * 16 + v + 8 * hi) * AST + col] = acc[v];
    }

    __syncthreads();

    // ---- phase 2: pair expansion + bias + relu, coalesced writes ----
    float* ob = out + (size_t)b * OUT_PER_B;
    for (int c = 0; c < Cn; ++c) {
        const float bc = bias[c];
        float* orow = ob + c * Pn;
        for (int p = tid; p < Pn; p += 256) {
            const float v = As[0][iiL[p] * AST + c]
                          + As[1][jjL[p] * AST + c] + bc;
            orow[p] = fmaxf(v, 0.0f);
        }
    }
}

extern "C" void kernel_launch(void* const* d_in, const int* in_sizes, int n_in,
                              void* d_out, int out_size, void* d_ws, size_t ws_size,
                              hipStream_t stream) {
    const float* x    = (const float*)d_in[0];  // x_emb (B,F,E) f32
    const float* W    = (const float*)d_in[1];  // W (C,1,2,E) f32
    const float* bias = (const float*)d_in[2];  // b (C,) f32
    float* out        = (float*)d_out;          // (B, C, P) reinterpreted flat
    interconv_kernel<<<Bn, 256, 0, stream>>>(x, W, bias, out);
}